// DualBranchModel_45775761441086
// MI455X (gfx1250) — compile-verified
//
#include <hip/hip_runtime.h>
#include <hip/hip_bf16.h>

// Problem constants (match reference)
static constexpr int B_   = 256;
static constexpr int NROW = 196;   // H*W
static constexpr int C_   = 768;
static constexpr int HID_ = 512;
static constexpr int NC_  = 701;
static constexpr float EPS_ = 1e-5f;

typedef float v2f __attribute__((ext_vector_type(2)));
typedef float v8f __attribute__((ext_vector_type(8)));

// ---------------------------------------------------------------------------
// Kernel 1: M[branch][b][n] = sum_c x[b][n][c]   (first full sweep of x1/x2)
// One 256-thread block per row; C_ = 3*256 so each thread loads 3 floats.
// ---------------------------------------------------------------------------
__global__ void rowsum_kernel(const float* __restrict__ x1,
                              const float* __restrict__ x2,
                              float* __restrict__ Mbuf)
{
    int id = blockIdx.x;                       // [0, 2*B_*NROW)
    int branch = id / (B_ * NROW);
    int rem = id - branch * (B_ * NROW);       // b*NROW + n
    const float* x = branch ? x2 : x1;
    const float* row = x + (size_t)rem * C_;
    int tid = threadIdx.x;
    float s = row[tid] + row[tid + 256] + row[tid + 512];
    __shared__ float red[256];
    red[tid] = s;
    __syncthreads();
    for (int st = 128; st > 0; st >>= 1) {
        if (tid < st) red[tid] += red[tid + st];
        __syncthreads();
    }
    if (tid == 0) Mbuf[id] = red[0];
}

// ---------------------------------------------------------------------------
// Kernel 2: per (branch, b): normalize scores, stable rank, gap-argmax -> k.
// Stores rank[n] and k. One block of 256 threads per (branch,b).
// ---------------------------------------------------------------------------
__global__ void sgm_select_kernel(const float* __restrict__ Mbuf,
                                  int* __restrict__ rankbuf,
                                  int* __restrict__ karr)
{
    int blk = blockIdx.x;                      // branch*B_ + b
    const float* Mrow = Mbuf + (size_t)blk * NROW;
    __shared__ float mnorm[NROW];
    __shared__ float sorted[NROW];
    __shared__ float red[256];
    __shared__ float mn_s, mx_s;
    int tid = threadIdx.x;

    float v = (tid < NROW) ? Mrow[tid] : 0.0f;

    // min reduce
    red[tid] = (tid < NROW) ? v : 3.4e38f;
    __syncthreads();
    for (int st = 128; st > 0; st >>= 1) {
        if (tid < st) red[tid] = fminf(red[tid], red[tid + st]);
        __syncthreads();
    }
    if (tid == 0) mn_s = red[0];
    __syncthreads();
    // max reduce
    red[tid] = (tid < NROW) ? v : -3.4e38f;
    __syncthreads();
    for (int st = 128; st > 0; st >>= 1) {
        if (tid < st) red[tid] = fmaxf(red[tid], red[tid + st]);
        __syncthreads();
    }
    if (tid == 0) mx_s = red[0];
    __syncthreads();

    float range = mx_s - mn_s;
    float inv = (range != 0.0f) ? (1.0f / range) : 0.0f;
    if (tid < NROW) mnorm[tid] = (v - mn_s) * inv;
    __syncthreads();

    // Stable rank: rank = #{j : m[j] < m[i]} + #{j < i : m[j] == m[i]}
    if (tid < NROW) {
        float mi = mnorm[tid];
        int r = 0;
        for (int j = 0; j < NROW; ++j) {
            float mj = mnorm[j];
            r += (mj < mi) || (mj == mi && j < tid);
        }
        sorted[r] = mi;
        rankbuf[(size_t)blk * NROW + tid] = r;
    }
    __syncthreads();

    // diff[r] = (sorted[r]==0) ? 0 : sorted[r+1]-sorted[r]; k = first argmax
    if (tid == 0) {
        float best = -3.4e38f;
        int k = 0;
        for (int r = 0; r < NROW - 1; ++r) {
            float left = sorted[r];
            float d = (left == 0.0f) ? 0.0f : (sorted[r + 1] - left);
            if (d > best) { best = d; k = r; }
        }
        karr[blk] = k;
    }
}

// ---------------------------------------------------------------------------
// Kernel 3: bg/fg masked means (second full sweep of x1/x2).
// Grid (2*B_, 3); thread -> one channel c; loop over the 196 rows.
// Output X4 layout: [x1_bg, x1_fg, x2_bg, x2_fg] each [B_, C_].
// ---------------------------------------------------------------------------
__global__ void sgm_accum_kernel(const float* __restrict__ x1,
                                 const float* __restrict__ x2,
                                 const int* __restrict__ rankbuf,
                                 const int* __restrict__ karr,
                                 float* __restrict__ X4)
{
    int blk = blockIdx.x;                      // branch*B_ + b
    int branch = blk >> 8;
    int b = blk & 255;
    int c = blockIdx.y * 256 + threadIdx.x;    // [0, C_)
    const float* x = branch ? x2 : x1;

    __shared__ int rs[NROW];
    __shared__ int ks;
    if (threadIdx.x < NROW) rs[threadIdx.x] = rankbuf[(size_t)blk * NROW + threadIdx.x];
    if (threadIdx.x == 0) ks = karr[blk];
    __syncthreads();
    int k = ks;

    const float* base = x + ((size_t)b * NROW) * C_ + c;
    float bg = 0.0f, fg = 0.0f;
    for (int n = 0; n < NROW; ++n) {
        float val = base[(size_t)n * C_];
        if (rs[n] < k) bg += val; else fg += val;
    }
    float kb = (float)((k > 1) ? k : 1);
    float kf = (float)(((NROW - k) > 1) ? (NROW - k) : 1);
    X4[((size_t)(2 * branch)     * B_ + b) * C_ + c] = bg / kb;
    X4[((size_t)(2 * branch + 1) * B_ + b) * C_ + c] = fg / kf;
}

// ---------------------------------------------------------------------------
// Kernel 4/6: fp32 GEMM + bias via V_WMMA_F32_16X16X4_F32.
// One wave (32 threads) computes one 16x16 tile of D = A[ M x K ] * Bm[ K x N ] + bias.
// A layout (ISA 7.12.2, 32-bit A 16x4): lanes 0-15 hold K=0,1; lanes 16-31 K=2,3.
// C/D layout: VGPR v -> row v (lanes 0-15) / v+8 (lanes 16-31), col = lane&15.
// M is a multiple of 16 (1024); N guarded (701 case); K multiple of 4.
// ---------------------------------------------------------------------------
__global__ void gemm_bias_wmma(const float* __restrict__ A,
                               const float* __restrict__ Bm,
                               const float* __restrict__ bias,
                               float* __restrict__ D,
                               int N, int K, int lda, int ldb, int ldd)
{
    int lane = threadIdx.x;                    // 0..31, wave32
    int half = lane >> 4;                      // 0 or 1
    int lm = lane & 15;
    int row = blockIdx.x * 16 + lm;            // A row this lane streams
    int col = blockIdx.y * 16 + lm;            // B/D column this lane owns
    int colc = (col < N) ? col : (N - 1);      // clamped (avoid OOB)
    float cmask = (col < N) ? 1.0f : 0.0f;

    v8f acc = {};
    const float* arow = A + (size_t)row * lda;
    for (int k0 = 0; k0 < K; k0 += 4) {
        int ka = k0 + half * 2;
        v2f a, b;
        a.x = arow[ka];
        a.y = arow[ka + 1];
        b.x = Bm[(size_t)(ka)     * ldb + colc] * cmask;
        b.y = Bm[(size_t)(ka + 1) * ldb + colc] * cmask;
        acc = __builtin_amdgcn_wmma_f32_16x16x4_f32(
            /*neg_a=*/false, a, /*neg_b=*/false, b,
            /*c_mod=*/(short)0, acc, /*reuse_a=*/false, /*reuse_b=*/false);
    }

    if (col < N) {
        float bv = bias[col];
        int rbase = blockIdx.x * 16 + half * 8;
#pragma unroll
        for (int v = 0; v < 8; ++v) {
            D[(size_t)(rbase + v) * ldd + col] = acc[v] + bv;
        }
    }
}

// ---------------------------------------------------------------------------
// Kernel 5: training-mode BN (biased var) + ReLU + dropout mask, in place on H.
// One block per classifier i (4 blocks), 512 threads = one column each.
// ---------------------------------------------------------------------------
__global__ void bn_relu_drop_kernel(float* __restrict__ H,
                                    const float* __restrict__ gamma,
                                    const float* __restrict__ beta,
                                    const float* __restrict__ dropmask)
{
    int i = blockIdx.x;                        // 0..3
    int j = threadIdx.x;                       // 0..511
    float* Hi = H + (size_t)i * B_ * HID_;
    const float* dm = dropmask + (size_t)i * B_ * HID_;

    float s = 0.0f, ss = 0.0f;
    for (int b = 0; b < B_; ++b) {
        float v = Hi[(size_t)b * HID_ + j];
        s += v; ss += v * v;
    }
    float mu = s * (1.0f / B_);
    float var = ss * (1.0f / B_) - mu * mu;    // biased, matches jnp.var / torch BN
    float scale = gamma[j] * rsqrtf(var + EPS_);
    float shift = beta[j] - mu * scale;

    for (int b = 0; b < B_; ++b) {
        float v = Hi[(size_t)b * HID_ + j] * scale + shift;
        v = fmaxf(v, 0.0f);
        v = (dm[(size_t)b * HID_ + j] > 0.5f) ? (v * 2.0f) : 0.0f;
        Hi[(size_t)b * HID_ + j] = v;
    }
}

// ---------------------------------------------------------------------------
// Kernel 7: softmax over axis 0 (the batch axis), per (i, class) column,
// in place on d_out [4][B_][NC_].
// ---------------------------------------------------------------------------
__global__ void softmax_dim0_kernel(float* __restrict__ out)
{
    int idx = blockIdx.x * blockDim.x + threadIdx.x;
    if (idx >= 4 * NC_) return;
    int i = idx / NC_;
    int j = idx - i * NC_;
    float* colp = out + (size_t)i * B_ * NC_ + j;

    float mx = -3.4e38f;
    for (int b = 0; b < B_; ++b) mx = fmaxf(mx, colp[(size_t)b * NC_]);
    float s = 0.0f;
    for (int b = 0; b < B_; ++b) s += expf(colp[(size_t)b * NC_] - mx);
    float invs = 1.0f / s;
    for (int b = 0; b < B_; ++b) {
        colp[(size_t)b * NC_] = expf(colp[(size_t)b * NC_] - mx) * invs;
    }
}

// ---------------------------------------------------------------------------
extern "C" void kernel_launch(void* const* d_in, const int* in_sizes, int n_in,
                              void* d_out, int out_size, void* d_ws, size_t ws_size,
                              hipStream_t stream)
{
    const float* x1       = (const float*)d_in[0];
    const float* x2       = (const float*)d_in[1];
    const float* W1       = (const float*)d_in[2];
    const float* b1       = (const float*)d_in[3];
    const float* gamma    = (const float*)d_in[4];
    const float* beta     = (const float*)d_in[5];
    const float* W2       = (const float*)d_in[6];
    const float* b2       = (const float*)d_in[7];
    const float* dropmask = (const float*)d_in[8];
    float* out = (float*)d_out;

    // Workspace carve-up (~6.05 MB total)
    float* Mbuf   = (float*)d_ws;                          // 2*B_*NROW floats
    int*   rankb  = (int*)(Mbuf + 2 * B_ * NROW);          // 2*B_*NROW ints
    int*   karr   = rankb + 2 * B_ * NROW;                 // 2*B_ ints
    float* X4     = (float*)(karr + 2 * B_);               // 4*B_*C_ floats
    float* Hbuf   = X4 + 4 * B_ * C_;                      // 4*B_*HID_ floats

    // 1) row sums (sweep 1 over x1/x2)
    rowsum_kernel<<<2 * B_ * NROW, 256, 0, stream>>>(x1, x2, Mbuf);

    // 2) sort/split-point per (branch, batch)
    sgm_select_kernel<<<2 * B_, 256, 0, stream>>>(Mbuf, rankb, karr);

    // 3) masked bg/fg means (sweep 2 over x1/x2)
    sgm_accum_kernel<<<dim3(2 * B_, C_ / 256), 256, 0, stream>>>(x1, x2, rankb, karr, X4);

    // 4) GEMM1: [4*B_ x C_] @ [C_ x HID_] + b1 -> H, fp32 WMMA
    gemm_bias_wmma<<<dim3((4 * B_) / 16, HID_ / 16), 32, 0, stream>>>(
        X4, W1, b1, Hbuf, HID_, C_, C_, HID_, HID_);

    // 5) BN (training stats) + ReLU + dropout, in place
    bn_relu_drop_kernel<<<4, HID_, 0, stream>>>(Hbuf, gamma, beta, dropmask);

    // 6) GEMM2: [4*B_ x HID_] @ [HID_ x NC_] + b2 -> logits in d_out (col-guarded)
    gemm_bias_wmma<<<dim3((4 * B_) / 16, (NC_ + 15) / 16), 32, 0, stream>>>(
        Hbuf, W2, b2, out, NC_, HID_, HID_, NC_, NC_);

    // 7) softmax over batch axis, in place on d_out
    softmax_dim0_kernel<<<(4 * NC_ + 255) / 256, 256, 0, stream>>>(out);
}